// GraphAttentionConvolution_75814762709297
// MI455X (gfx1250) — compile-verified
//
#include <hip/hip_runtime.h>
#include <hip/hip_bf16.h>

typedef float v2f __attribute__((ext_vector_type(2)));
typedef float v8f __attribute__((ext_vector_type(8)));
typedef unsigned int u32x4 __attribute__((ext_vector_type(4)));
typedef int i32x8 __attribute__((ext_vector_type(8)));
typedef int i32x4 __attribute__((ext_vector_type(4)));

__device__ __forceinline__ v8f wmma4(v2f a, v2f b, v8f c) {
  // V_WMMA_F32_16X16X4_F32: full fp32 precision matrix op (wave32)
  return __builtin_amdgcn_wmma_f32_16x16x4_f32(false, a, false, b, (short)0, c,
                                               false, false);
}

#define DFEAT 256

// -------------------------------------------------------------------------
// TDM: DMA a [tile_rows x 256] fp32 tile (row-major, stride 256) into LDS.
// D# packed per CDNA5 ISA §8.3/8.4 (group0: count/lds_addr/global_addr/type,
// group1: data_size=4B, tensor_dim0=256, dim0_stride=256, tile 256 x rows).
// amdgpu-toolchain (clang-23) 6-arg builtin form.
// -------------------------------------------------------------------------
__device__ __forceinline__ void tdm_load_rows(const float* src, unsigned lds_off,
                                              unsigned rows_total, int tile_rows) {
  unsigned long long ga = (unsigned long long)(const void*)src;
  u32x4 g0;
  g0[0] = 1u;                                   // count=1 (valid user D#)
  g0[1] = lds_off;                              // lds_addr (bytes)
  g0[2] = (unsigned)ga;                         // global_addr[31:0]
  g0[3] = (unsigned)((ga >> 32) & 0x1FFFFFFull) // global_addr[56:32]
          | (2u << 30);                         // type = 2 ("image")
  i32x8 g1;
  g1[0] = (2 << 16);                            // data_size = 4 bytes
  g1[1] = (int)(256u << 16);                    // tensor_dim0 = 256 (bits 79:48)
  g1[2] = (int)((rows_total & 0xFFFFu) << 16);  // tensor_dim1 lo16 (bits 111:80)
  g1[3] = (int)(256u << 16)                     // tile_dim0 = 256 (bits 127:112)
          | (int)(rows_total >> 16);            // tensor_dim1 hi16
  g1[4] = tile_rows;                            // tile_dim1 (bits 143:128)
  g1[5] = 256;                                  // tensor_dim0_stride (bits 207:160)
  g1[6] = 0;
  g1[7] = 0;                                    // tensor_dim1_stride unused (2D)
  i32x4 z4 = {0, 0, 0, 0};
  i32x8 z8 = {0, 0, 0, 0, 0, 0, 0, 0};
  __builtin_amdgcn_tensor_load_to_lds(g0, g1, z4, z4, z8, 0);
}

// -------------------------------------------------------------------------
// Kernel 1: h[b] = x[b] @ W   (fp32 WMMA, 64x64 tile per 128-thread block)
// grid: (DFEAT/64, N/64, B), block: 128 (4 waves, 2x2 wave grid, 32x32 each)
// -------------------------------------------------------------------------
__global__ void __launch_bounds__(128)
gemm_f32_wmma(const float* __restrict__ X, const float* __restrict__ W,
              float* __restrict__ H, int N, int K) {
  __shared__ float As[64][20];  // pad 4: A-frag reads hit distinct banks
  __shared__ float Bs[16][68];

  const int tid  = threadIdx.x;
  const int lane = tid & 31;
  const int wave = tid >> 5;
  const int wr   = wave >> 1, wc = wave & 1;
  const int n0   = blockIdx.x * 64;   // feature (D) offset
  const int row0 = blockIdx.y * 64;   // row (N) offset
  const int b    = blockIdx.z;

  const float* Xb = X + (size_t)b * N * K;
  float*       Hb = H + (size_t)b * N * DFEAT;

  const int m  = lane & 15;   // matrix row/col within 16
  const int kh = lane >> 4;   // K half-select (lanes 16-31 hold K+2)

  v8f acc[2][2] = {};

  for (int k0 = 0; k0 < K; k0 += 16) {
    // stage A tile 64x16 (float4 coalesced)
    for (int idx = tid; idx < 256; idx += 128) {
      int r = idx >> 2, kv = (idx & 3) * 4;
      float4 v = *(const float4*)(Xb + (size_t)(row0 + r) * K + k0 + kv);
      As[r][kv + 0] = v.x; As[r][kv + 1] = v.y;
      As[r][kv + 2] = v.z; As[r][kv + 3] = v.w;
    }
    // stage B tile 16x64
    for (int idx = tid; idx < 256; idx += 128) {
      int kk = idx >> 4, cv = (idx & 15) * 4;
      float4 v = *(const float4*)(W + (size_t)(k0 + kk) * DFEAT + n0 + cv);
      Bs[kk][cv + 0] = v.x; Bs[kk][cv + 1] = v.y;
      Bs[kk][cv + 2] = v.z; Bs[kk][cv + 3] = v.w;
    }
    __syncthreads();

    for (int kk = 0; kk < 16; kk += 4) {
      v2f a0, a1, b0, b1;
      // A 16x4 layout: VGPR0 = {K0 | K2}, VGPR1 = {K1 | K3}
      a0.x = As[wr * 32 +      m][kk + 2 * kh];
      a0.y = As[wr * 32 +      m][kk + 2 * kh + 1];
      a1.x = As[wr * 32 + 16 + m][kk + 2 * kh];
      a1.y = As[wr * 32 + 16 + m][kk + 2 * kh + 1];
      // B 4x16 layout mirrors A
      b0.x = Bs[kk + 2 * kh    ][wc * 32 +      m];
      b0.y = Bs[kk + 2 * kh + 1][wc * 32 +      m];
      b1.x = Bs[kk + 2 * kh    ][wc * 32 + 16 + m];
      b1.y = Bs[kk + 2 * kh + 1][wc * 32 + 16 + m];
      acc[0][0] = wmma4(a0, b0, acc[0][0]);
      acc[0][1] = wmma4(a0, b1, acc[0][1]);
      acc[1][0] = wmma4(a1, b0, acc[1][0]);
      acc[1][1] = wmma4(a1, b1, acc[1][1]);
    }
    __syncthreads();
  }

  // C/D layout: VGPR r holds rows {r | r+8}
  for (int i = 0; i < 2; ++i)
    for (int j = 0; j < 2; ++j)
      for (int r = 0; r < 8; ++r) {
        int gm = row0 + wr * 32 + i * 16 + r + 8 * kh;
        int gn = n0 + wc * 32 + j * 16 + m;
        Hb[(size_t)gm * DFEAT + gn] = acc[i][j][r];
      }
}

// -------------------------------------------------------------------------
// Kernel 2: three fused row-dot-products f_k[row] = h[row,:] . a_k[:]
// one wave per row; 8 rows per 256-thread block
// -------------------------------------------------------------------------
__global__ void __launch_bounds__(256)
rowdot3(const float* __restrict__ H, const float* __restrict__ A1,
        const float* __restrict__ A2, const float* __restrict__ A3,
        float* __restrict__ F1, float* __restrict__ F2,
        float* __restrict__ F3, int nrows) {
  const int lane = threadIdx.x & 31;
  const int wave = threadIdx.x >> 5;
  const int row  = blockIdx.x * 8 + wave;
  if (row >= nrows) return;
  const float* h = H + (size_t)row * DFEAT;
  float s1 = 0.f, s2 = 0.f, s3 = 0.f;
  for (int d = lane; d < DFEAT; d += 32) {
    float x = h[d];
    s1 += x * A1[d]; s2 += x * A2[d]; s3 += x * A3[d];
  }
  for (int s = 16; s > 0; s >>= 1) {
    s1 += __shfl_xor(s1, s, 32);
    s2 += __shfl_xor(s2, s, 32);
    s3 += __shfl_xor(s3, s, 32);
  }
  if (lane == 0) { F1[row] = s1; F2[row] = s2; F3[row] = s3; }
}

// -------------------------------------------------------------------------
// Kernel 3: fused masked softmax attention + (att*deg) @ g + bias
// grid: (N/32, B), block 256 (8 waves). 32-row tile (halves g L2 traffic vs
// 16-row). LDS: P[32][M] score tile + 2x(64x256) TDM-staged g chunks,
// double-buffered with tensor_load_to_lds + s_wait_tensorcnt.
// -------------------------------------------------------------------------
#define PSTRIDE 1028          // 1024 + 4 pad: A-frag LDS reads conflict-free
#define KC      64            // g chunk rows per TDM transfer (64KB/chunk)

__global__ void __launch_bounds__(256)
gat_fused(const float* __restrict__ G, const int* __restrict__ Adj,
          const float* __restrict__ E1, const float* __restrict__ E2,
          const float* __restrict__ Bias, float* __restrict__ Out,
          int N, int M) {
  extern __shared__ float smem[];
  float* P        = smem;                    // [32][PSTRIDE]
  float* rowscale = smem + 32 * PSTRIDE;     // [32]
  float* gbuf     = rowscale + 32;           // [2][KC*256]

  const int tid  = threadIdx.x;
  const int lane = tid & 31;
  const int wave = tid >> 5;
  const int row0 = blockIdx.x * 32;
  const int b    = blockIdx.y;

  const float* Gb  = G  + (size_t)b * M * DFEAT;
  const float* e2b = E2 + (size_t)b * M;
  const int nchunks = M / KC;

  // ---- Pre-issue the first two g-chunk DMAs; they overlap Phase A ----
  const unsigned gbuf_base = (unsigned)(unsigned long long)gbuf;
  if (wave == 0) {
    tdm_load_rows(Gb, gbuf_base, (unsigned)M, KC);
    tdm_load_rows(Gb + (size_t)KC * DFEAT,
                  gbuf_base + KC * DFEAT * 4u, (unsigned)(M - KC), KC);
  }

  // ---- Phase A: masked scores + softmax stats (4 rows per wave) ----
  for (int rr = 0; rr < 4; ++rr) {
    const int r   = wave * 4 + rr;
    const int row = row0 + r;
    const int* adjrow = Adj + ((size_t)b * N + row) * M;
    const float e1 = E1[(size_t)b * N + row];
    float* Pr = P + (size_t)r * PSTRIDE;

    float vmax = -3.0e38f, deg = 0.f;
    for (int c = lane; c < M; c += 32) {
      __builtin_prefetch(adjrow + c + 256, 0, 0);  // global_prefetch_b8
      int   a = adjrow[c];
      float e = e1 + e2b[c];
      e = (e > 0.f) ? e : 0.2f * e;          // leaky_relu(0.2)
      float v = (a > 0) ? e : -9.0e15f;      // adjacency mask
      Pr[c] = v;
      vmax  = fmaxf(vmax, v);
      deg  += (float)a;
    }
    for (int s = 16; s > 0; s >>= 1) {
      vmax = fmaxf(vmax, __shfl_xor(vmax, s, 32));
      deg += __shfl_xor(deg, s, 32);
    }
    float lsum = 0.f;
    for (int c = lane; c < M; c += 32) {
      float p = __expf(Pr[c] - vmax);        // masked -> exp(-huge) = 0
      Pr[c] = p;
      lsum += p;
    }
    for (int s = 16; s > 0; s >>= 1) lsum += __shfl_xor(lsum, s, 32);
    // deg==0 row: softmax uniform but att*deg == 0 -> scale 0 (matches ref)
    if (lane == 0) rowscale[r] = (deg > 0.f) ? (deg / lsum) : 0.f;
  }

  // ---- Phase B: out[32,256] = P[32,M] @ g[M,256]; wave owns 32 cols ----
  const int m  = lane & 15;
  const int kh = lane >> 4;
  const int n0 = wave * 32;
  const float* Pm0 = P + (size_t)m * PSTRIDE;
  const float* Pm1 = P + (size_t)(16 + m) * PSTRIDE;

  v8f acc[2][2] = {};
  for (int c = 0; c < nchunks; ++c) {
    if (wave == 0) {
      // steady state: 2 outstanding, chunk c done when cnt<=1; tail: cnt<=0
      if (c < nchunks - 1) __builtin_amdgcn_s_wait_tensorcnt(1);
      else                 __builtin_amdgcn_s_wait_tensorcnt(0);
    }
    __syncthreads();   // publish DMA'd chunk (and, on c==0, P/rowscale)

    const float* gs = gbuf + (size_t)(c & 1) * (KC * DFEAT);
    const int kc = c * KC;
    for (int km = 0; km < KC; km += 4) {
      v2f a0 = *(const v2f*)(Pm0 + kc + km + 2 * kh);   // ds_load_b64
      v2f a1 = *(const v2f*)(Pm1 + kc + km + 2 * kh);
      const float* g0 = gs + (size_t)(km + 2 * kh) * DFEAT;
      const float* g1 = g0 + DFEAT;
      v2f b0, b1;
      b0.x = g0[n0 + m];      b0.y = g1[n0 + m];
      b1.x = g0[n0 + 16 + m]; b1.y = g1[n0 + 16 + m];
      acc[0][0] = wmma4(a0, b0, acc[0][0]);
      acc[0][1] = wmma4(a0, b1, acc[0][1]);
      acc[1][0] = wmma4(a1, b0, acc[1][0]);
      acc[1][1] = wmma4(a1, b1, acc[1][1]);
    }
    __syncthreads();   // all waves done with gbuf[c&1] before re-filling it
    if (wave == 0 && c + 2 < nchunks) {
      tdm_load_rows(Gb + (size_t)(c + 2) * KC * DFEAT,
                    gbuf_base + (unsigned)(c & 1) * (KC * DFEAT * 4u),
                    (unsigned)(M - (c + 2) * KC), KC);
    }
  }

  float* Ob = Out + ((size_t)b * N + row0) * DFEAT;
  for (int i = 0; i < 2; ++i)
    for (int j = 0; j < 2; ++j)
      for (int r = 0; r < 8; ++r) {
        int gm = i * 16 + r + 8 * kh;      // C/D layout: VGPR r = rows r|r+8
        int gn = n0 + j * 16 + m;
        Ob[(size_t)gm * DFEAT + gn] = rowscale[gm] * acc[i][j][r] + Bias[gn];
      }
}

// -------------------------------------------------------------------------
extern "C" void kernel_launch(void* const* d_in, const int* in_sizes, int n_in,
                              void* d_out, int out_size, void* d_ws,
                              size_t ws_size, hipStream_t stream) {
  (void)in_sizes; (void)n_in; (void)out_size; (void)ws_size;

  const float* x0   = (const float*)d_in[0];
  const float* x1   = (const float*)d_in[1];
  const float* W0   = (const float*)d_in[2];
  const float* W1   = (const float*)d_in[3];
  const float* a1_0 = (const float*)d_in[4];
  const float* a2_0 = (const float*)d_in[5];
  const float* a1_1 = (const float*)d_in[6];
  const float* a2_1 = (const float*)d_in[7];
  const float* bias = (const float*)d_in[8];
  const int* adj00  = (const int*)d_in[9];
  const int* adj01  = (const int*)d_in[10];
  const int* adj10  = (const int*)d_in[11];
  const int* adj11  = (const int*)d_in[12];

  const int B = 64, N1 = 1024, N2 = 512, F0 = 512, F1 = 384;

  float* ws = (float*)d_ws;
  float* h0    = ws;                            // 64*1024*256
  float* h1    = h0 + (size_t)B * N1 * DFEAT;   // 64*512*256
  float* f0a1  = h1 + (size_t)B * N2 * DFEAT;   // h0 @ a1_0   [B*N1]
  float* f0a20 = f0a1  + (size_t)B * N1;        // h0 @ a2_0
  float* f0a21 = f0a20 + (size_t)B * N1;        // h0 @ a2_1
  float* f1a1  = f0a21 + (size_t)B * N1;        // h1 @ a1_1   [B*N2]
  float* f1a20 = f1a1  + (size_t)B * N2;        // h1 @ a2_0
  float* f1a21 = f1a20 + (size_t)B * N2;        // h1 @ a2_1

  float* out = (float*)d_out;
  float* o00 = out;
  float* o01 = o00 + (size_t)B * N1 * DFEAT;
  float* o10 = o01 + (size_t)B * N1 * DFEAT;
  float* o11 = o10 + (size_t)B * N2 * DFEAT;

  // h = x @ W (fp32 WMMA)
  gemm_f32_wmma<<<dim3(DFEAT / 64, N1 / 64, B), 128, 0, stream>>>(x0, W0, h0, N1, F0);
  gemm_f32_wmma<<<dim3(DFEAT / 64, N2 / 64, B), 128, 0, stream>>>(x1, W1, h1, N2, F1);

  // attention score vectors (single pass over each h)
  rowdot3<<<(B * N1) / 8, 256, 0, stream>>>(h0, a1_0, a2_0, a2_1,
                                            f0a1, f0a20, f0a21, B * N1);
  rowdot3<<<(B * N2) / 8, 256, 0, stream>>>(h1, a1_1, a2_0, a2_1,
                                            f1a1, f1a20, f1a21, B * N2);

  const size_t smemB =
      (size_t)(32 * PSTRIDE + 32 + 2 * KC * DFEAT) * sizeof(float); // ~257KB
  // o00 = att(h0,h0,adj00,a1_0,a2_0)
  gat_fused<<<dim3(N1 / 32, B), 256, smemB, stream>>>(h0, adj00, f0a1, f0a20, bias, o00, N1, N1);
  // o01 = att(h0,h1,adj01,a1_0,a2_0)
  gat_fused<<<dim3(N1 / 32, B), 256, smemB, stream>>>(h1, adj01, f0a1, f1a20, bias, o01, N1, N2);
  // o10 = att(h1,h0,adj10,a1_1,a2_1)
  gat_fused<<<dim3(N2 / 32, B), 256, smemB, stream>>>(h0, adj10, f1a1, f0a21, bias, o10, N2, N1);
  // o11 = att(h1,h1,adj11,a1_1,a2_1)
  gat_fused<<<dim3(N2 / 32, B), 256, smemB, stream>>>(h1, adj11, f1a1, f1a21, bias, o11, N2, N2);
}